// DSAM_27315992002865
// MI455X (gfx1250) — compile-verified
//
#include <hip/hip_runtime.h>
#include <hip/hip_bf16.h>
#include <math.h>

// ---------------- problem constants ----------------
constexpr int B = 16;
constexpr int N = 2048;
constexpr int D = 512;
constexpr int K = 8;

#define TAU 0.07f
#define EPSv 1e-6f
#define LN_EPS 1e-5f
#define NORM_EPS 1e-12f

// ---------------- output layout (flat, return order) ----------------
constexpr size_t OFF_FFG  = 0;                          // [B,N,D]
constexpr size_t OFF_FBG  = OFF_FFG + (size_t)B*N*D;    // [B,N,D]
constexpr size_t OFF_PBG  = OFF_FBG + (size_t)B*N*D;    // [B,K,D]
constexpr size_t OFF_SSEM = OFF_PBG + (size_t)B*K*D;    // [B,N]
constexpr size_t OFF_LC   = OFF_SSEM + (size_t)B*N;     // scalar
constexpr size_t OFF_LO   = OFF_LC + 1;                 // scalar
constexpr size_t OFF_LA   = OFF_LO + 1;                 // scalar

// ---------------- workspace layout (floats) ----------------
constexpr int WS_COMPACT = 0;   // sum of min(1-cos)
constexpr int WS_OFULL   = 1;   // sum_b ||P^T P||_F^2
constexpr int WS_ODIAG   = 2;   // sum_{b,n} (||p_n||^2)^2
constexpr int WS_ALIGN   = 3;   // sum neg_log * mask
constexpr int WS_CNT     = 4;   // sum mask
constexpr int WS_DENOM   = 8;                    // [B*K]
constexpr int WS_INVP    = WS_DENOM + B*K;       // [B*K] 1/max(||p_bg||,eps)
constexpr int WS_INVBG   = WS_INVP + B*K;        // [B*N] 1/max(||f_bg||,eps)

typedef float  v2f __attribute__((ext_vector_type(2)));
typedef float  v8f __attribute__((ext_vector_type(8)));

__device__ __forceinline__ float wave_sum(float v) {
#pragma unroll
    for (int o = 16; o > 0; o >>= 1) v += __shfl_xor(v, o, 32);
    return v;
}

// ---------------- kernel 0: zero scalar accumulators ----------------
__global__ void dsam_init(float* ws) {
    if (threadIdx.x < 8) ws[threadIdx.x] = 0.0f;
}

// ---------------- kernel 1: per-(b,k) attention softmax -> p_bg ----------------
// grid = B*K blocks, 256 threads (8 waves)
__global__ void dsam_proto(const float* __restrict__ f_raw,
                           const float* __restrict__ q_bg,
                           float* __restrict__ out, float* __restrict__ ws) {
    __shared__ float logit[N];      // 8 KB
    __shared__ float red[256];

    const int bk = blockIdx.x;
    const int b  = bk / K;
    const int k  = bk % K;
    const int tid  = threadIdx.x;
    const int wave = tid >> 5;
    const int lane = tid & 31;

    const float* Fb = f_raw + (size_t)b * N * D;
    const float* q  = q_bg + (size_t)k * D;
    const float inv_sqrt_d = 0.044194173824159216f;   // 1/sqrt(512)

    // phase 1: logits (one wave per token per iteration)
    for (int n = wave; n < N; n += 8) {
        const float* fr = Fb + (size_t)n * D;
        float p = 0.0f;
#pragma unroll 4
        for (int d = lane; d < D; d += 32) p = fmaf(fr[d], q[d], p);
        p = wave_sum(p);
        if (lane == 0) logit[n] = p * inv_sqrt_d;
    }
    __syncthreads();

    // block max
    float m = -3.4e38f;
    for (int i = tid; i < N; i += 256) m = fmaxf(m, logit[i]);
    red[tid] = m; __syncthreads();
    for (int s = 128; s > 0; s >>= 1) {
        if (tid < s) red[tid] = fmaxf(red[tid], red[tid + s]);
        __syncthreads();
    }
    const float mx = red[0];
    __syncthreads();

    // exp + block sum
    float ssum = 0.0f;
    for (int i = tid; i < N; i += 256) {
        float e = __expf(logit[i] - mx);
        logit[i] = e;
        ssum += e;
    }
    red[tid] = ssum; __syncthreads();
    for (int s = 128; s > 0; s >>= 1) {
        if (tid < s) red[tid] += red[tid + s];
        __syncthreads();
    }
    const float inv_sum = 1.0f / red[0];
    __syncthreads();
    for (int i = tid; i < N; i += 256) logit[i] *= inv_sum;
    __syncthreads();

    // phase 2: p_bg[b,k,d] = sum_n w[n] * f_raw[b,n,d]
    const int d0 = tid, d1 = tid + 256;
    float acc0 = 0.0f, acc1 = 0.0f;
    for (int n = 0; n < N; ++n) {
        const float w = logit[n];
        const float* fr = Fb + (size_t)n * D;
        acc0 = fmaf(w, fr[d0], acc0);
        acc1 = fmaf(w, fr[d1], acc1);
    }
    float* pb = out + OFF_PBG + ((size_t)b * K + k) * D;
    pb[d0] = acc0;
    pb[d1] = acc1;

    // denom = ||p||^2 + eps, invp = 1/max(||p||, 1e-12)
    red[tid] = acc0 * acc0 + acc1 * acc1; __syncthreads();
    for (int s = 128; s > 0; s >>= 1) {
        if (tid < s) red[tid] += red[tid + s];
        __syncthreads();
    }
    if (tid == 0) {
        const float p2 = red[0];
        ws[WS_DENOM + b * K + k] = p2 + EPSv;
        ws[WS_INVP  + b * K + k] = 1.0f / fmaxf(sqrtf(p2), NORM_EPS);
    }
}

// ---------------- kernel 2: fused per-token pipeline ----------------
// one wave32 per token; 8 tokens per block; grid = B*(N/8)
__global__ void dsam_token(const float* __restrict__ f_raw,
                           const float* __restrict__ f_q,
                           const int*   __restrict__ gt_mask,
                           const float* __restrict__ alpha_p,
                           const float* __restrict__ ln_w,
                           const float* __restrict__ ln_b,
                           float* __restrict__ out, float* __restrict__ ws) {
    __shared__ float p_s[K * D];    // 16 KB prototypes
    __shared__ float fq_s[D];       // 2 KB
    __shared__ float den_s[K], ivp_s[K];
    __shared__ float acc_s[4];

    const int blocksPerB = N / 8;
    const int b    = blockIdx.x / blocksPerB;
    const int tile = blockIdx.x % blocksPerB;
    const int tid  = threadIdx.x;
    const int wave = tid >> 5;
    const int lane = tid & 31;

    for (int i = tid; i < K * D; i += 256) p_s[i] = out[OFF_PBG + (size_t)b * K * D + i];
    for (int i = tid; i < D; i += 256)     fq_s[i] = f_q[(size_t)b * D + i];
    if (tid < K) { den_s[tid] = ws[WS_DENOM + b * K + tid];
                   ivp_s[tid] = ws[WS_INVP  + b * K + tid]; }
    if (tid < 4) acc_s[tid] = 0.0f;
    __syncthreads();

    const int n = tile * 8 + wave;
    const float* fr = f_raw + ((size_t)b * N + n) * D;

    float fv[16];
#pragma unroll
    for (int j = 0; j < 16; ++j) fv[j] = fr[lane + 32 * j];

    float raw2 = 0.0f;
#pragma unroll
    for (int j = 0; j < 16; ++j) raw2 = fmaf(fv[j], fv[j], raw2);
    raw2 = wave_sum(raw2);

    float num[K];
#pragma unroll
    for (int k = 0; k < K; ++k) {
        float t = 0.0f;
#pragma unroll
        for (int j = 0; j < 16; ++j) t = fmaf(fv[j], p_s[k * D + lane + 32 * j], t);
        num[k] = wave_sum(t);
    }

    const float alpha = alpha_p[0];
    float coef[K];
#pragma unroll
    for (int k = 0; k < K; ++k) coef[k] = num[k] / den_s[k];

    float xb[16], bgv[16];
    float sx = 0.0f, sx2 = 0.0f, bg2 = 0.0f;
#pragma unroll
    for (int j = 0; j < 16; ++j) {
        float bg = 0.0f;
#pragma unroll
        for (int k = 0; k < K; ++k) bg = fmaf(coef[k], p_s[k * D + lane + 32 * j], bg);
        bgv[j] = bg;
        const float x = fv[j] - alpha * bg;
        xb[j] = x;
        sx  += x;
        sx2 = fmaf(x, x, sx2);
        bg2 = fmaf(bg, bg, bg2);
    }
    sx = wave_sum(sx); sx2 = wave_sum(sx2); bg2 = wave_sum(bg2);

    const float mu   = sx * (1.0f / D);
    const float var  = fmaxf(sx2 * (1.0f / D) - mu * mu, 0.0f);
    const float rstd = rsqrtf(var + LN_EPS);

    float fg[16];
    float fg2 = 0.0f, sdot = 0.0f, fq2 = 0.0f;
#pragma unroll
    for (int j = 0; j < 16; ++j) {
        const int d = lane + 32 * j;
        const float g = fmaf((xb[j] - mu) * rstd, ln_w[d], ln_b[d]);
        fg[j] = g;
        fg2  = fmaf(g, g, fg2);
        const float qv = fq_s[d];
        sdot = fmaf(g, qv, sdot);
        fq2  = fmaf(qv, qv, fq2);
    }
    fg2 = wave_sum(fg2); sdot = wave_sum(sdot); fq2 = wave_sum(fq2);

    const float inv_fg = 1.0f / fmaxf(sqrtf(fg2), NORM_EPS);
    const float inv_q  = 1.0f / fmaxf(sqrtf(fq2), NORM_EPS);
    const float sim    = sdot * inv_fg * inv_q;
    const float ssem   = 1.0f / (1.0f + __expf(-sim / TAU));

    const size_t base = ((size_t)b * N + n) * D;
#pragma unroll
    for (int j = 0; j < 16; ++j) {
        out[OFF_FFG + base + lane + 32 * j] = fg[j];
        out[OFF_FBG + base + lane + 32 * j] = bgv[j];
    }

    if (lane == 0) {
        out[OFF_SSEM + (size_t)b * N + n] = ssem;

        const float inv_raw = 1.0f / fmaxf(sqrtf(raw2), NORM_EPS);
        float mind = 3.4e38f;
#pragma unroll
        for (int k = 0; k < K; ++k) {
            const float dk = num[k] * inv_raw * ivp_s[k];
            mind = fminf(mind, 1.0f - dk);
        }
        const float invbg = 1.0f / fmaxf(sqrtf(bg2), NORM_EPS);
        ws[WS_INVBG + b * N + n] = invbg;
        const float pn2  = bg2 * invbg * invbg;     // ||p_n||^2 (diag of gram)
        const float mval = (gt_mask[(size_t)b * N + n] != 0) ? 1.0f : 0.0f;
        const float nl   = -__logf(fmaxf(ssem, EPSv));

        atomicAdd(&acc_s[0], mind);
        atomicAdd(&acc_s[1], pn2 * pn2);
        atomicAdd(&acc_s[2], nl * mval);
        atomicAdd(&acc_s[3], mval);
    }
    __syncthreads();
    if (tid == 0) {
        atomicAdd(&ws[WS_COMPACT], acc_s[0]);
        atomicAdd(&ws[WS_ODIAG],   acc_s[1]);
        atomicAdd(&ws[WS_ALIGN],   acc_s[2]);
        atomicAdd(&ws[WS_CNT],     acc_s[3]);
    }
}

// ---------------- kernel 3: loss_orth via ||P^T P||_F^2, fp32 WMMA ----------------
// P = l2norm(f_bg) in [N, D].  C = P^T P in [D, D];  accumulate sum(C^2).
// Block computes a 64x64 region of C (4x4 tiles of 16x16; 8 waves x 2 tiles).
// grid = B * (512/64)^2 = 1024 blocks, 256 threads.
#define GTILE 64
#define NCH   32
__global__ void dsam_gram(const float* __restrict__ out_ro,
                          float* __restrict__ ws) {
    __shared__ float As[NCH][GTILE + 1];
    __shared__ float Bs[NCH][GTILE + 1];
    __shared__ float blk_acc;

    const int b  = blockIdx.x >> 6;          // 64 region-blocks per batch
    const int t  = blockIdx.x & 63;
    const int bi = (t >> 3) * GTILE;
    const int bj = (t & 7)  * GTILE;

    const float* P   = out_ro + OFF_FBG + (size_t)b * N * D;
    const float* ivn = ws + WS_INVBG + b * N;

    const int tid  = threadIdx.x;
    const int wave = tid >> 5;
    const int lane = tid & 31;

    if (tid == 0) blk_acc = 0.0f;
    __syncthreads();

    // tile assignment: wave handles tiles t0 = 2*wave, t1 = 2*wave+1 (same ti)
    const int ti  = (wave * 2) >> 2;        // 0..3
    const int i0  = ti * 16;
    const int tj0 = ((wave * 2) & 3) * 16;
    const int tj1 = ((wave * 2 + 1) & 3) * 16;

    v8f c0 = {}; v8f c1 = {};

    const int cw = tid & 63;                // column within 64
    const int rw = tid >> 6;                // 0..3

    for (int n0 = 0; n0 < N; n0 += NCH) {
        // stage normalized rows into LDS
        for (int r = rw; r < NCH; r += 4) {
            const int n = n0 + r;
            const float iv = ivn[n];
            As[r][cw] = P[(size_t)n * D + bi + cw] * iv;
            Bs[r][cw] = P[(size_t)n * D + bj + cw] * iv;
        }
        if (n0 + NCH < N) {
            __builtin_prefetch(&P[(size_t)(n0 + NCH + rw) * D + bi + cw], 0, 0);
            __builtin_prefetch(&P[(size_t)(n0 + NCH + rw) * D + bj + cw], 0, 0);
        }
        __syncthreads();

        // WMMA over this n-chunk: 16x16x4 fp32
        const int m  = lane & 15;
        const int kh = (lane < 16) ? 0 : 2;  // lanes 16-31 hold K=2,3
#pragma unroll
        for (int kk = 0; kk < NCH; kk += 4) {
            const int kb = kk + kh;
            v2f a;  a.x  = As[kb][i0 + m];   a.y  = As[kb + 1][i0 + m];
            v2f b0; b0.x = Bs[kb][tj0 + m];  b0.y = Bs[kb + 1][tj0 + m];
            v2f b1; b1.x = Bs[kb][tj1 + m];  b1.y = Bs[kb + 1][tj1 + m];
            c0 = __builtin_amdgcn_wmma_f32_16x16x4_f32(
                    false, a, false, b0, (short)0, c0, false, false);
            c1 = __builtin_amdgcn_wmma_f32_16x16x4_f32(
                    false, a, false, b1, (short)0, c1, false, false);
        }
        __syncthreads();
    }

    float s = 0.0f;
#pragma unroll
    for (int e = 0; e < 8; ++e) s += c0[e] * c0[e] + c1[e] * c1[e];
    atomicAdd(&blk_acc, s);
    __syncthreads();
    if (tid == 0) atomicAdd(&ws[WS_OFULL], blk_acc);
}

// ---------------- kernel 4: finalize the three scalar losses ----------------
__global__ void dsam_finalize(float* __restrict__ out, const float* __restrict__ ws) {
    if (threadIdx.x == 0) {
        out[OFF_LC] = ws[WS_COMPACT] / (float)((size_t)B * N);
        out[OFF_LO] = (ws[WS_OFULL] - ws[WS_ODIAG]) / ((float)B * (float)N * (float)N);
        const float cnt = ws[WS_CNT];
        out[OFF_LA] = (cnt > 0.0f) ? ws[WS_ALIGN] / fmaxf(cnt, 1.0f) : 0.0f;
    }
}

// ---------------- launcher ----------------
extern "C" void kernel_launch(void* const* d_in, const int* in_sizes, int n_in,
                              void* d_out, int out_size, void* d_ws, size_t ws_size,
                              hipStream_t stream) {
    const float* f_raw  = (const float*)d_in[0];
    const float* f_q    = (const float*)d_in[1];
    const int*   gt     = (const int*)  d_in[2];
    const float* q_bg   = (const float*)d_in[3];
    const float* alpha  = (const float*)d_in[4];
    const float* ln_w   = (const float*)d_in[5];
    const float* ln_b   = (const float*)d_in[6];
    float* out = (float*)d_out;
    float* ws  = (float*)d_ws;

    dsam_init<<<1, 32, 0, stream>>>(ws);
    dsam_proto<<<B * K, 256, 0, stream>>>(f_raw, q_bg, out, ws);
    dsam_token<<<B * (N / 8), 256, 0, stream>>>(f_raw, f_q, gt, alpha, ln_w, ln_b, out, ws);
    dsam_gram<<<B * 64, 256, 0, stream>>>(out, ws);
    dsam_finalize<<<1, 32, 0, stream>>>(out, ws);
}